// PhysicsInformedFeatureExtractor_13297218748849
// MI455X (gfx1250) — compile-verified
//
#include <hip/hip_runtime.h>

// Problem constants (from reference): B=4, N=50000, F=128, E=1600000
#define NB 4
#define NN 50000
#define NF 128
#define NE 1600000

// Native clang vector types: required by __builtin_nontemporal_store
// (HIP's float2/float4 are structs and are rejected by the builtin).
typedef float f32x2 __attribute__((ext_vector_type(2)));
typedef float f32x4 __attribute__((ext_vector_type(4)));

// ---------------------------------------------------------------------------
// Kernel 1: compact voltages (node_features[:,:,0]) into an L2-resident table
//           and zero the per-node accumulators.
//           init maxload = 0 is exact: reference does max(segment_max, 0).
// ---------------------------------------------------------------------------
__global__ __launch_bounds__(256) void pife_init_kernel(
    const float* __restrict__ nf,
    float* __restrict__ volt,
    float* __restrict__ maxload,
    float* __restrict__ stabsum)
{
    int i = blockIdx.x * blockDim.x + threadIdx.x;
    if (i < NB * NN) {
        volt[i]    = nf[(size_t)i * NF];   // strided read, done once
        maxload[i] = 0.0f;
        stabsum[i] = 0.0f;
    }
}

// ---------------------------------------------------------------------------
// Kernel 2: per-edge scatter. One thread per edge, unrolled over batches so
//           the src/dst index load is amortized 4x. Voltage gathers hit the
//           800KB compacted table (L2-resident on the 192MB L2).
//           Edge streams (lf/tl/sus) are read exactly once -> non-temporal
//           loads keep them from evicting the hot tables.
//           loading >= 0 always (lf in [0,1), tl+1e-6 > 0), so float max ==
//           signed-int atomic max on the bit pattern.
// ---------------------------------------------------------------------------
__global__ __launch_bounds__(256) void pife_edge_kernel(
    const int*   __restrict__ ei,    // [2, E]
    const float* __restrict__ lf,    // [B, E]
    const float* __restrict__ tl,    // [B, E]
    const float* __restrict__ sus,   // [B, E]
    const float* __restrict__ volt,  // [B, N]
    float* __restrict__ maxload,     // [B, N]
    float* __restrict__ stabsum)     // [B, N]
{
    int e = blockIdx.x * blockDim.x + threadIdx.x;
    if (e >= NE) return;
    int s = ei[e];
    int d = ei[NE + e];
#pragma unroll
    for (int b = 0; b < NB; ++b) {
        size_t be  = (size_t)b * NE + e;
        int   bn_s = b * NN + s;
        int   bn_d = b * NN + d;

        float flow = __builtin_nontemporal_load(&lf[be]);
        float lim  = __builtin_nontemporal_load(&tl[be]);
        float bsus = __builtin_nontemporal_load(&sus[be]);

        float loading = flow / (lim + 1e-6f);
        atomicMax((int*)&maxload[bn_s], __float_as_int(loading)); // nonneg floats

        float vi  = volt[bn_s];
        float vj  = volt[bn_d];
        float fij = bsus * vj / (vi + 1e-6f);
        atomicAdd(&stabsum[bn_s], fij);                           // global_atomic_add_f32
    }
}

// ---------------------------------------------------------------------------
// Kernel 3: per-node output assembly. One wave32 handles TWO adjacent
//           128-float rows, double-buffered through the CDNA5 async
//           global->LDS engine:
//             issue async(row0), async(row1)          (ASYNCcnt -> 2)
//             s_wait_asynccnt 1  -> row0 landed (async loads complete
//                                   in order per ISA) ; store row0
//             s_wait_asynccnt 0  -> row1 landed        ; store row1
//           Output rows are 130 floats (520B stride: 8B-aligned, not 16B),
//           so stores are b64 pairs, marked non-temporal (write-once data).
// ---------------------------------------------------------------------------
__global__ __launch_bounds__(256) void pife_node_kernel(
    const float* __restrict__ nf,       // [B, N, F]
    const float* __restrict__ maxload,  // [B, N]
    const float* __restrict__ stabsum,  // [B, N]
    float* __restrict__ out)            // [B, N, F+2]
{
    __shared__ float tile[8 * 2 * NF];  // 8 waves * 2 rows * 512B = 8KB LDS

    const int lane = threadIdx.x & 31;
    const int wave = threadIdx.x >> 5;
    const int row0 = (blockIdx.x * 8 + wave) * 2;  // grid sized exactly: no guard
    const int row1 = row0 + 1;

    float* buf0 = &tile[(wave * 2 + 0) * NF];
    float* buf1 = &tile[(wave * 2 + 1) * NF];

    // Each lane moves 16B; a wave moves one full 512B row per instruction.
    unsigned long long ga0 =
        (unsigned long long)(uintptr_t)(nf + (size_t)row0 * NF + lane * 4);
    unsigned long long ga1 =
        (unsigned long long)(uintptr_t)(nf + (size_t)row1 * NF + lane * 4);
    // Generic LDS pointer -> architected 32-bit LDS byte offset (addr[31:0]).
    unsigned lo0 = (unsigned)(uintptr_t)(buf0 + lane * 4);
    unsigned lo1 = (unsigned)(uintptr_t)(buf1 + lane * 4);

    asm volatile("global_load_async_to_lds_b128 %0, %1, off"
                 :: "v"(lo0), "v"(ga0) : "memory");
    asm volatile("global_load_async_to_lds_b128 %0, %1, off"
                 :: "v"(lo1), "v"(ga1) : "memory");

    // ---- consume row0 while row1 is still in flight ----
    asm volatile("s_wait_asynccnt 0x1" ::: "memory");
    {
        const f32x4 v = *(const f32x4*)(buf0 + lane * 4);
        float* orow = out + (size_t)row0 * (NF + 2);
        __builtin_nontemporal_store((f32x2){v.x, v.y}, (f32x2*)orow + lane * 2);
        __builtin_nontemporal_store((f32x2){v.z, v.w}, (f32x2*)orow + lane * 2 + 1);
        if (lane == 0) {
            float ml = maxload[row0];                 // already >= 0 (init 0)
            f32x2 extra = {ml > 1.0f ? 1.0f : 0.0f,
                           fabsf(1.0f - stabsum[row0])};
            __builtin_nontemporal_store(extra, (f32x2*)(orow + NF));
        }
    }

    // ---- consume row1 ----
    asm volatile("s_wait_asynccnt 0x0" ::: "memory");
    {
        const f32x4 v = *(const f32x4*)(buf1 + lane * 4);
        float* orow = out + (size_t)row1 * (NF + 2);
        __builtin_nontemporal_store((f32x2){v.x, v.y}, (f32x2*)orow + lane * 2);
        __builtin_nontemporal_store((f32x2){v.z, v.w}, (f32x2*)orow + lane * 2 + 1);
        if (lane == 0) {
            float ml = maxload[row1];
            f32x2 extra = {ml > 1.0f ? 1.0f : 0.0f,
                           fabsf(1.0f - stabsum[row1])};
            __builtin_nontemporal_store(extra, (f32x2*)(orow + NF));
        }
    }
}

// ---------------------------------------------------------------------------
// Launcher
// ---------------------------------------------------------------------------
extern "C" void kernel_launch(void* const* d_in, const int* in_sizes, int n_in,
                              void* d_out, int out_size, void* d_ws, size_t ws_size,
                              hipStream_t stream)
{
    const float* nf  = (const float*)d_in[0];  // node_features  [B,N,F]
    const int*   ei  = (const int*)  d_in[1];  // edge_index     [2,E]
    const float* lf  = (const float*)d_in[2];  // line_flows     [B,E,1]
    const float* tl  = (const float*)d_in[3];  // thermal_limits [B,E,1]
    const float* sus = (const float*)d_in[4];  // susceptance    [B,E,1]
    float* out = (float*)d_out;                // [B,N,F+2]

    // Workspace: 3 * B*N floats = 2.4 MB
    float* volt    = (float*)d_ws;
    float* maxload = volt    + NB * NN;
    float* stabsum = maxload + NB * NN;

    const int tpb = 256;
    hipLaunchKernelGGL(pife_init_kernel, dim3((NB * NN + tpb - 1) / tpb), dim3(tpb),
                       0, stream, nf, volt, maxload, stabsum);
    hipLaunchKernelGGL(pife_edge_kernel, dim3((NE + tpb - 1) / tpb), dim3(tpb),
                       0, stream, ei, lf, tl, sus, volt, maxload, stabsum);
    // 8 waves/block * 2 rows/wave = 16 rows/block; 200000/16 = 12500 blocks
    hipLaunchKernelGGL(pife_node_kernel, dim3((NB * NN) / 16), dim3(tpb),
                       0, stream, nf, maxload, stabsum, out);
}